// RandomTimeStretch_56856777065174
// MI455X (gfx1250) — compile-verified
//
#include <hip/hip_runtime.h>
#include <math.h>

// Phase vocoder (torchaudio-style), rate=1.1, hop=512 on (16, 1025, 2048) f32 STFT.
//
// Strategy: one 256-thread block (8 wave32) per 16-row tile.
//   * Per 16-time x 16-row chunk: phase tile (B operand) -> inclusive prefix sum
//     D = L(16x16 lower-tri ones) x B via four chained V_WMMA_F32_16X16X4_F32.
//   * 8 waves process 8 consecutive chunks in parallel; cross-chunk carries
//     resolved with a tiny LDS exchange per round (8x16 totals + double-buffered
//     per-row base), 2 barriers/round.
//   * Transcendentals on the VALU/TRANS pipes (v_rcp/v_sin/v_cos), prefix sums
//     on the matrix pipe.

#define T_IN     2048
#define T_OUT    1862               // len(np.arange(0, 2048, 1.1))
#define NFREQ    1025
#define NBATCH   16
#define NROWS    (NBATCH * NFREQ)   // 16400 flat (b, f) rows; 16400/16 = 1025 tiles
#define RATE_D   1.1
#define WAVES    8                  // waves per block (chunks in flight per tile)
#define TWO_PI_F 6.28318530717958647692f
#define INV_2PI  0.15915494309189533577f  // 0x3E22F983 (hw inline constant)
#define PI_F     3.14159265358979323846f
#define HALFPI_F 1.57079632679489661923f

typedef __attribute__((ext_vector_type(2))) float v2f;
typedef __attribute__((ext_vector_type(8))) float v8f;

// Branch-free atan2 built on v_rcp_f32 + degree-11 odd minimax poly on [0,1]
// (max err ~1.7e-6 rad). Quadrant fixup via selects; sign via copysign.
__device__ __forceinline__ float fast_atan2f(float y, float x) {
  float ax = __builtin_fabsf(x);
  float ay = __builtin_fabsf(y);
  float mx = fmaxf(ax, ay);
  float mn = fminf(ax, ay);
  float r  = mn * __builtin_amdgcn_rcpf(mx);     // ratio in [0,1]
  r = (mx == 0.0f) ? 0.0f : r;                   // atan2(0,0) -> 0 like the ref
  float s = r * r;
  float p =             -0.01172120f;
  p = __builtin_fmaf(p, s, 0.05265332f);
  p = __builtin_fmaf(p, s, -0.11643287f);
  p = __builtin_fmaf(p, s, 0.19354346f);
  p = __builtin_fmaf(p, s, -0.33262347f);
  p = __builtin_fmaf(p, s, 0.99997726f);
  float a = p * r;                               // atan(mn/mx) in [0, pi/4+]
  a = (ay > ax) ? (HALFPI_F - a) : a;            // true angle vs x-axis
  a = (x < 0.0f) ? (PI_F - a) : a;               // quadrant 2/3
  return __builtin_copysignf(a, y);              // sign of y (handles -0)
}

// sin/cos of the accumulated phase. phase_acc reaches ~3e6 where f32 ulp is
// 0.25 rad -- the reference's own f32 cumsum is noise mod 2pi there, so a
// cheap mod-2pi reduction + hardware trig (V_SIN_F32/V_COS_F32 take input in
// revolutions: D = sin(S0*2pi), domain +-256) is the right cost/accuracy point.
__device__ __forceinline__ void fast_sincos(float pacc, float* sn, float* cs) {
  float q  = pacc * INV_2PI;       // revolutions
  float qr = q - rintf(q);         // [-0.5, 0.5] revolutions == pacc mod 2pi
  *sn = __builtin_amdgcn_sinf(qr);
  *cs = __builtin_amdgcn_cosf(qr);
}

// phase[t] as defined by the reference AFTER the concat/shift:
//   t == 0 : angle0 at input frame 0
//   t >= 1 : wrap(angle1 - angle0 - pa) + pa evaluated at output step t-1
__device__ __forceinline__ float phase_term(const float* __restrict__ re,
                                            const float* __restrict__ im,
                                            long rowbase, int t, float pa) {
  if (t == 0) {
    return fast_atan2f(im[rowbase], re[rowbase]);
  }
  double s = (double)(t - 1) * RATE_D;   // matches np.arange(0, T, rate)[t-1]
  int i0 = (int)s;                       // <= 2046 for t-1 <= 1860 -> i0+1 in range
  float r0 = re[rowbase + i0],     m0 = im[rowbase + i0];
  float r1 = re[rowbase + i0 + 1], m1 = im[rowbase + i0 + 1];
  float a0 = fast_atan2f(m0, r0);
  float a1 = fast_atan2f(m1, r1);
  float ph = a1 - a0 - pa;
  // round-half-even == jnp.round; 1/(2pi) multiply instead of f32 divide
  ph = ph - TWO_PI_F * rintf(ph * INV_2PI);
  return ph + pa;
}

// mag[t] = alpha*|frame(idx+1)| + (1-alpha)*|frame(idx)|; idx+1 may hit zero pad.
// Branchless: clamp the load, select 0 for the padded frame.
__device__ __forceinline__ float mag_term(const float* __restrict__ re,
                                          const float* __restrict__ im,
                                          long rowbase, int t) {
  double s = (double)t * RATE_D;
  int i0 = (int)s;
  float alpha = (float)(s - (double)i0);
  float r0 = re[rowbase + i0], m0 = im[rowbase + i0];
  float n0 = __builtin_amdgcn_sqrtf(r0 * r0 + m0 * m0);
  int i1 = i0 + 1;
  int i1c = (i1 < T_IN) ? i1 : (T_IN - 1);
  float r1 = re[rowbase + i1c], m1 = im[rowbase + i1c];
  float n1 = (i1 < T_IN) ? __builtin_amdgcn_sqrtf(r1 * r1 + m1 * m1) : 0.0f;
  return alpha * n1 + (1.0f - alpha) * n0;
}

__global__ __launch_bounds__(32 * WAVES)
void phase_vocoder_wmma(const float* __restrict__ re,
                        const float* __restrict__ im,
                        float* __restrict__ out_re,
                        float* __restrict__ out_im) {
  const int  w    = threadIdx.x >> 5;              // wave id in block: 0..7
  const int  lane = threadIdx.x & 31;
  const bool hi   = lane >= 16;
  const int  n    = lane & 15;                     // row-in-tile (A's M, B/D's N)
  const long grow = (long)blockIdx.x * 16 + n;     // global flat (b,f) row
  const long rowbase = grow * (long)T_IN;
  const int  f = (int)(grow % NFREQ);
  // linspace(0, pi*hop, n_freq)[f] = pi*512*f/1024 = (pi/2)*f
  const float pa = HALFPI_F * (float)f;

  // Per-round chunk totals (written by hi lanes: D[M=15][n]) and double-buffered
  // per-row running carry.
  __shared__ float s_tot[WAVES][16];
  __shared__ float s_base[2][16];
  if (threadIdx.x < 16) {
    s_base[0][threadIdx.x] = 0.0f;
  }

  // A = 16x16 lower-triangular ones, as four 16x4 K-slices.
  // A layout (32-bit, 16x4): lane m = lane&15; VGPR{0,1} hold K = {0,1} (lo lanes)
  // or K = {2,3} (hi lanes).
  v2f A[4];
#pragma unroll
  for (int k = 0; k < 4; ++k) {
    int kk = 4 * k + (hi ? 2 : 0);
    A[k].x = (kk     <= n) ? 1.0f : 0.0f;
    A[k].y = (kk + 1 <= n) ? 1.0f : 0.0f;
  }

  const int nchunks = (T_OUT + 15) / 16;               // 117 (last partial: 6 valid)
  const int nrounds = (nchunks + WAVES - 1) / WAVES;   // 15

  for (int r = 0; r < nrounds; ++r) {
    const int  c     = r * WAVES + w;
    const bool valid = (c < nchunks);
    const int  tbase = c * 16;

    v8f acc = {0.f, 0.f, 0.f, 0.f, 0.f, 0.f, 0.f, 0.f};
    if (valid) {
      // Prefetch the input window of this wave's chunk in the NEXT round.
      {
        int tn = tbase + 16 * WAVES;
        if (tn < T_OUT) {
          int ip = (int)((double)tn * RATE_D);
          __builtin_prefetch(&re[rowbase + ip], 0, 1);
          __builtin_prefetch(&im[rowbase + ip], 0, 1);
        }
      }
      // B = phase tile, B[K=time][N=row], as four 4x16 K-slices.
      // B layout (32-bit, 4x16): lane holds N = lane&15; VGPR{0,1} hold
      // K = {4k, 4k+1} (lo lanes) or K = {4k+2, 4k+3} (hi lanes).
      v2f B[4];
#pragma unroll
      for (int k = 0; k < 4; ++k) {
        int t0 = tbase + 4 * k + (hi ? 2 : 0);
        B[k].x = (t0     < T_OUT) ? phase_term(re, im, rowbase, t0,     pa) : 0.0f;
        B[k].y = (t0 + 1 < T_OUT) ? phase_term(re, im, rowbase, t0 + 1, pa) : 0.0f;
      }
      // D = L x B : inclusive prefix over 16 time steps for all 16 rows.
#pragma unroll
      for (int k = 0; k < 4; ++k) {
        acc = __builtin_amdgcn_wmma_f32_16x16x4_f32(
            /*neg_a=*/false, A[k], /*neg_b=*/false, B[k],
            /*c_mod=*/(short)0, acc, /*reuse_a=*/false, /*reuse_b=*/false);
      }
    }

    // Chunk total for row n is D[M=15][N=n] = VGPR7 of hi lane n (0 if invalid).
    if (hi) {
      s_tot[w][n] = acc[7];
    }
    __syncthreads();   // totals (this round) + base (prev round) visible

    // Exclusive carry for this wave's chunk: base + totals of earlier waves.
    // Fully unrolled, predicated (no exec-mask inner loop).
    float off = s_base[r & 1][n];
#pragma unroll
    for (int wp = 0; wp < WAVES - 1; ++wp) {
      float tv = s_tot[wp][n];
      off += (wp < w) ? tv : 0.0f;
    }

    // Last valid wave publishes the next round's base (double buffer).
    int cmax = nchunks - 1 - r * WAVES;
    if (cmax > WAVES - 1) cmax = WAVES - 1;
    if (w == cmax && hi) {
      s_base[(r + 1) & 1][n] = off + s_tot[w][n];
    }

    if (valid) {
      // D layout: lane n (lo) holds times tbase+j, lane n+16 holds tbase+8+j.
#pragma unroll
      for (int j = 0; j < 8; ++j) {
        int tg = tbase + j + (hi ? 8 : 0);
        if (tg < T_OUT) {
          float pacc = off + acc[j];
          float mg = mag_term(re, im, rowbase, tg);
          float sn, cs;
          fast_sincos(pacc, &sn, &cs);
          long o = grow * (long)T_OUT + tg;
          out_re[o] = mg * cs;
          out_im[o] = mg * sn;
        }
      }
    }
    __syncthreads();   // next base visible; s_tot reusable next round
  }
}

extern "C" void kernel_launch(void* const* d_in, const int* in_sizes, int n_in,
                              void* d_out, int out_size, void* d_ws, size_t ws_size,
                              hipStream_t stream) {
  (void)in_sizes; (void)n_in; (void)out_size; (void)d_ws; (void)ws_size;
  const float* re = (const float*)d_in[0];
  const float* im = (const float*)d_in[1];
  float* out_re = (float*)d_out;
  float* out_im = out_re + (long)NROWS * (long)T_OUT;   // tuple concat: re then im
  dim3 grid(NROWS / 16);     // 1025 tiles of 16 rows
  dim3 block(32 * WAVES);    // 8 wave32 per tile, chained-scan pipeline
  phase_vocoder_wmma<<<grid, block, 0, stream>>>(re, im, out_re, out_im);
}